// MultiHeadSelfAttention_7945689497641
// MI455X (gfx1250) — compile-verified
//
#include <hip/hip_runtime.h>

// Problem constants (from reference)
#define EMBED 1024
#define HEADS 16
#define HDIM  64
#define BATCH 8
#define SEQ   2048
#define MROWS (BATCH * SEQ)   // 16384

// Workspace layout (floats)
//   wvsum : [1024 x 64]    at 0
//   wosum : [64 x 1024]    at 65536
//   weff  : [1024 x 1024]  at 131072
#define WS_WVSUM 0
#define WS_WOSUM 65536
#define WS_WEFF  131072

typedef __attribute__((ext_vector_type(2))) float v2f;
typedef __attribute__((ext_vector_type(8))) float v8f;

// ---------------------------------------------------------------------------
// Prep 1: Wv_sum[e][d] = sum_h Wv[e][h*64+d]   (1024 x 64)
// ---------------------------------------------------------------------------
__global__ void reduce_wv_kernel(const float* __restrict__ Wv,
                                 float* __restrict__ wvsum) {
    int idx = blockIdx.x * blockDim.x + threadIdx.x;   // 65536 total
    int e = idx >> 6;
    int d = idx & 63;
    float s = 0.0f;
    const float* p = Wv + e * EMBED + d;
#pragma unroll
    for (int h = 0; h < HEADS; ++h) s += p[h * HDIM];
    wvsum[idx] = s;
}

// ---------------------------------------------------------------------------
// Prep 2: Wo_sum[d][o] = sum_a Wo[a*64+d][o]   (64 x 1024)
// ---------------------------------------------------------------------------
__global__ void reduce_wo_kernel(const float* __restrict__ Wo,
                                 float* __restrict__ wosum) {
    int idx = blockIdx.x * blockDim.x + threadIdx.x;   // 65536 total
    int d = idx >> 10;
    int o = idx & 1023;
    float s = 0.0f;
    const float* p = Wo + d * EMBED + o;
#pragma unroll
    for (int a = 0; a < HEADS; ++a) s += p[a * HDIM * EMBED];
    wosum[idx] = s;
}

// ---------------------------------------------------------------------------
// Prep 3: W_eff[k][n] = sum_d Wv_sum[k][d] * Wo_sum[d][n]   (1024 x 1024)
// 67 MFLOP total — plain FMA loop is plenty.
// ---------------------------------------------------------------------------
__global__ void weff_kernel(const float* __restrict__ wvsum,
                            const float* __restrict__ wosum,
                            float* __restrict__ weff) {
    int idx = blockIdx.x * blockDim.x + threadIdx.x;   // 1M total
    int k = idx >> 10;
    int n = idx & 1023;
    float s = 0.0f;
    const float* a = wvsum + k * HDIM;
    const float* b = wosum + n;
#pragma unroll
    for (int d = 0; d < HDIM; ++d) s = fmaf(a[d], b[d * EMBED], s);
    weff[idx] = s;
}

// ---------------------------------------------------------------------------
// Main GEMM: out[m][n] = sum_k x[m][k] * W_eff[k][n] + bo[n]
//   M = 16384, N = 1024, K = 1024, fp32 WMMA 16x16x4.
//   Block tile 128x128, BK = 32, 256 threads = 8 waves.
//   Wave (wm in 0..3, wn in 0..1) owns 32x64 = 2x4 WMMA 16x16 tiles.
// ---------------------------------------------------------------------------
#define BM 128
#define BN 128
#define BK 32
#define LDA 36    // 32 + 4 pad (keeps float4 alignment, breaks bank conflicts)
#define LDB 132   // 128 + 4 pad

__global__ __launch_bounds__(256)
void gemm_weff_kernel(const float* __restrict__ x,
                      const float* __restrict__ weff,
                      const float* __restrict__ bo,
                      float* __restrict__ out) {
    __shared__ float sA[BM * LDA];   // [128 rows][32 K]
    __shared__ float sB[BK * LDB];   // [32 K][128 cols]

    const int tid    = threadIdx.x;
    const int lane   = tid & 31;
    const int wave   = tid >> 5;          // 0..7
    const int wm     = wave >> 1;         // 0..3  -> 32 rows each
    const int wn     = wave & 1;          // 0..1  -> 64 cols each
    const int blockN = blockIdx.x * BN;
    const int blockM = blockIdx.y * BM;

    const int lhalf  = lane >> 4;         // 0 or 1
    const int l16    = lane & 15;

    // Accumulators: 2 (mi) x 4 (ni) tiles of 16x16 fp32
    v8f c[2][4];
#pragma unroll
    for (int mi = 0; mi < 2; ++mi)
#pragma unroll
        for (int ni = 0; ni < 4; ++ni)
            c[mi][ni] = (v8f)(0.0f);

    // Per-lane bias for each ni tile (col = blockN + wn*64 + ni*16 + l16)
    float bias[4];
#pragma unroll
    for (int ni = 0; ni < 4; ++ni)
        bias[ni] = bo[blockN + wn * 64 + ni * 16 + l16];

    const int KT = EMBED / BK;  // 32 K-tiles
    for (int kt = 0; kt < KT; ++kt) {
        // ---- stage A tile: 128 rows x 32 K = 1024 float4, 4 per thread ----
#pragma unroll
        for (int i = 0; i < 4; ++i) {
            int fid = tid + i * 256;
            int row = fid >> 3;              // 8 float4 per row
            int c4  = (fid & 7) << 2;
            float4 v = *(const float4*)(x + (size_t)(blockM + row) * EMBED
                                          + kt * BK + c4);
            *(float4*)(&sA[row * LDA + c4]) = v;
        }
        // ---- stage B tile: 32 K x 128 cols = 1024 float4, 4 per thread ----
#pragma unroll
        for (int i = 0; i < 4; ++i) {
            int fid = tid + i * 256;
            int row = fid >> 5;              // 32 float4 per row
            int c4  = (fid & 31) << 2;
            float4 v = *(const float4*)(weff + (size_t)(kt * BK + row) * EMBED
                                             + blockN + c4);
            *(float4*)(&sB[row * LDB + c4]) = v;
        }
        // Prefetch next K-tile into L2 while we compute (global_prefetch_b8)
        if (kt + 1 < KT) {
            __builtin_prefetch(x + (size_t)(blockM + (tid >> 1)) * EMBED
                                 + (kt + 1) * BK + ((tid & 1) << 4), 0, 0);
            __builtin_prefetch(weff + (size_t)((kt + 1) * BK + (tid >> 3)) * EMBED
                                    + blockN + ((tid & 7) << 4), 0, 0);
        }
        __syncthreads();

        // ---- 8 WMMA K-steps of 4 ----
#pragma unroll
        for (int ks = 0; ks < 8; ++ks) {
            const int kk = ks * 4 + lhalf * 2;   // this lane-half's K pair

            // A fragments: lane holds row (l16), K = kk..kk+1
            v2f a[2];
#pragma unroll
            for (int mi = 0; mi < 2; ++mi) {
                int rowA = wm * 32 + mi * 16 + l16;
                a[mi] = *(const v2f*)(&sA[rowA * LDA + kk]);
            }
            // B fragments: lane holds col (l16), K = kk..kk+1
            v2f b[4];
#pragma unroll
            for (int ni = 0; ni < 4; ++ni) {
                int colB = wn * 64 + ni * 16 + l16;
                v2f t;
                t.x = sB[kk * LDB + colB];
                t.y = sB[(kk + 1) * LDB + colB];
                b[ni] = t;
            }
#pragma unroll
            for (int mi = 0; mi < 2; ++mi)
#pragma unroll
                for (int ni = 0; ni < 4; ++ni)
                    c[mi][ni] = __builtin_amdgcn_wmma_f32_16x16x4_f32(
                        /*neg_a=*/false, a[mi],
                        /*neg_b=*/false, b[ni],
                        /*c_mod=*/(short)0, c[mi][ni],
                        /*reuse_a=*/false, /*reuse_b=*/false);
        }
        __syncthreads();
    }

    // ---- epilogue: bias add + store ----
    // C layout: VGPR r, lanes 0-15 -> row r, lanes 16-31 -> row r+8; col = l16
#pragma unroll
    for (int mi = 0; mi < 2; ++mi) {
#pragma unroll
        for (int ni = 0; ni < 4; ++ni) {
            const int col = blockN + wn * 64 + ni * 16 + l16;
#pragma unroll
            for (int r = 0; r < 8; ++r) {
                const int row = blockM + wm * 32 + mi * 16 + (lhalf << 3) + r;
                out[(size_t)row * EMBED + col] = c[mi][ni][r] + bias[ni];
            }
        }
    }
}

// ---------------------------------------------------------------------------
// Launch
// ---------------------------------------------------------------------------
extern "C" void kernel_launch(void* const* d_in, const int* in_sizes, int n_in,
                              void* d_out, int out_size, void* d_ws, size_t ws_size,
                              hipStream_t stream) {
    const float* x  = (const float*)d_in[0];
    // d_in[1] (Wq) and d_in[2] (Wk) are mathematically dead: softmax rows of
    // the (N,H,H) attention sum to 1 and are fully contracted away.
    const float* Wv = (const float*)d_in[3];
    const float* Wo = (const float*)d_in[4];
    const float* bo = (const float*)d_in[5];
    float* out = (float*)d_out;

    float* ws    = (float*)d_ws;
    float* wvsum = ws + WS_WVSUM;   // 1024 x 64
    float* wosum = ws + WS_WOSUM;   // 64 x 1024
    float* weff  = ws + WS_WEFF;    // 1024 x 1024

    reduce_wv_kernel<<<dim3(65536 / 256), dim3(256), 0, stream>>>(Wv, wvsum);
    reduce_wo_kernel<<<dim3(65536 / 256), dim3(256), 0, stream>>>(Wo, wosum);
    weff_kernel<<<dim3((1024 * 1024) / 256), dim3(256), 0, stream>>>(wvsum, wosum, weff);

    dim3 grid(EMBED / BN, MROWS / BM);   // (8, 128)
    gemm_weff_kernel<<<grid, dim3(256), 0, stream>>>(x, weff, bo, out);
}